// AvgClicksPoolingInitializer_28707561407357
// MI455X (gfx1250) — compile-verified
//
#include <hip/hip_runtime.h>

typedef float v2f __attribute__((ext_vector_type(2)));
typedef float v8f __attribute__((ext_vector_type(8)));

#define B_ 8
#define I_ 16
#define C_ 256

// ---------------- workspace layout (float offsets) ----------------
// sel level0: 8*16*16384 = 2097152 floats
// sel level1: 8*16*4096  =  524288
// sel level2: 8*16*1024  =  131072
// sel level3: 8*16*256   =   32768
// s accum   : 4 * 32768  =  131072
// cnt       : 4 * 128    =     512
// packed u64: 4 * 128 u64 (byte offset 2916864*4, 8-byte aligned)
static const size_t SEL_OFF[4] = {0, 2097152, 2621440, 2752512};
static const size_t S_OFF      = 2785280;
static const size_t CNT_OFF    = 2916352;
static const size_t PK_BYTEOFF = 2916864ull * 4ull;

// ---------------- zero accumulators ----------------
__global__ void zero_kernel(float* __restrict__ p, int n) {
    int i = blockIdx.x * blockDim.x + threadIdx.x;
    if (i < n) p[i] = 0.0f;
}

// ---------------- mask / cnt / argmax kernel ----------------
// One block of 256 threads handles 256 consecutive output pixels of one (b,i).
// Bilinear (antialias=False) downsample by integer factor f with half-pixel
// centers == 0.25 * 2x2 block at source offset f/2 - 1.
__global__ void __launch_bounds__(256)
mask_kernel(const float* __restrict__ scrib, float* __restrict__ sel,
            float* __restrict__ cnt, unsigned long long* __restrict__ packed,
            int w, int factor, int offs, int blocksPerBI) {
    __shared__ float scnt[256];
    __shared__ unsigned long long smax[256];
    int bi = blockIdx.x / blocksPerBI;        // b*16 + i
    int pb = blockIdx.x % blocksPerBI;
    int t  = threadIdx.x;
    int p  = pb * 256 + t;
    int y = p / w, x = p % w;
    int ys = y * factor + offs;
    int xs = x * factor + offs;
    const float* sp = scrib + (size_t)bi * 512 * 512;
    float m = 0.25f * (sp[(size_t)ys * 512 + xs]     + sp[(size_t)ys * 512 + xs + 1] +
                       sp[(size_t)(ys + 1) * 512 + xs] + sp[(size_t)(ys + 1) * 512 + xs + 1]);
    float sv = (m > 0.5f) ? 1.0f : 0.0f;       // FG_THRESH = 0.5, strict >
    sel[(size_t)bi * ((size_t)blocksPerBI * 256) + p] = sv;
    scnt[t] = sv;
    // m >= 0 always (avg of uniforms), so raw float bits are order-preserving.
    // Pack (m_bits << 32) | (~p) so atomicMax yields max m with smallest p (first-argmax).
    smax[t] = ((unsigned long long)__float_as_uint(m) << 32) |
              (unsigned long long)(0xFFFFFFFFu - (unsigned)p);
    __syncthreads();
    for (int st = 128; st > 0; st >>= 1) {
        if (t < st) {
            scnt[t] += scnt[t + st];
            if (smax[t + st] > smax[t]) smax[t] = smax[t + st];
        }
        __syncthreads();
    }
    if (t == 0) {
        atomicAdd(&cnt[bi], scnt[0]);
        atomicMax(&packed[bi], smax[0]);
    }
}

// ---------------- WMMA masked-sum einsum ----------------
// s[b,i,c] += sum_k sel[b,i,k] * feat[b,c,k]  via V_WMMA_F32_16X16X4_F32.
// One wave handles (b, 16-wide C tile, K chunk). M = I = 16, N = 16 of C, K = 4/step.
// A (16x4): lane l -> M = l&15, K = (l>>4)*2 .. +1  (2 VGPRs)
// B (4x16): lane l -> N = l&15, K = (l>>4)*2 .. +1  (2 VGPRs)
// D (16x16 f32, 8 VGPRs): vgpr r, lanes 0-15 -> M=r, lanes 16-31 -> M=8+r, N=lane&15.
__global__ void __launch_bounds__(256)
wmma_masked_sum_kernel(const float* __restrict__ feat, const float* __restrict__ sel,
                       float* __restrict__ s, int P, int chunk, int nchunks) {
    int wid  = (blockIdx.x * blockDim.x + threadIdx.x) >> 5;
    int lane = threadIdx.x & 31;
    int kc = wid % nchunks;
    int ct = (wid / nchunks) & 15;
    int b  = wid / (nchunks * 16);
    int k0 = kc * chunk;
    int row = lane & 15;             // M for A, N for B
    int kb  = (lane >> 4) * 2;       // K sub-offset within group of 4
    const float* ap = sel  + ((size_t)b * I_ + row) * (size_t)P + (size_t)(k0 + kb);
    const float* bp = feat + ((size_t)b * C_ + (size_t)ct * 16 + row) * (size_t)P + (size_t)(k0 + kb);
    v8f acc = {};
    for (int k = 0; k < chunk; k += 4) {
        v2f a  = *(const v2f*)(ap + k);
        v2f bv = *(const v2f*)(bp + k);
        // 8 args: (neg_a, A, neg_b, B, c_mod, C, reuse_a, reuse_b)
        acc = __builtin_amdgcn_wmma_f32_16x16x4_f32(false, a, false, bv,
                                                    (short)0, acc, false, false);
    }
    float* sd = s + ((size_t)b * I_ + (size_t)(lane >> 4) * 8) * C_ + (size_t)ct * 16 + (lane & 15);
#pragma unroll
    for (int r = 0; r < 8; ++r)
        atomicAdd(sd + (size_t)r * C_, acc[r]);
}

// ---------------- finalize: mean over levels with gather fallback ----------------
__global__ void __launch_bounds__(256)
finalize_kernel(const float* __restrict__ f0, const float* __restrict__ f1,
                const float* __restrict__ f2, const float* __restrict__ f3,
                const float* __restrict__ s, const float* __restrict__ cnt,
                const unsigned long long* __restrict__ packed, float* __restrict__ out) {
    int tid = blockIdx.x * blockDim.x + threadIdx.x;   // 0 .. B*I*C-1
    int c  = tid & 255;
    int bi = tid >> 8;
    int b  = bi >> 4;
    const float* feats[4] = {f0, f1, f2, f3};
    const int Ps[4] = {16384, 4096, 1024, 256};
    float sum = 0.0f;
#pragma unroll
    for (int l = 0; l < 4; ++l) {
        float cv = cnt[l * (B_ * I_) + bi];
        float v;
        if (cv > 0.0f) {
            v = s[l * (B_ * I_ * C_) + tid] / cv;      // max(cnt,1)==cnt when cnt>0
        } else {
            unsigned long long pk = packed[l * (B_ * I_) + bi];
            unsigned idx = 0xFFFFFFFFu - (unsigned)(pk & 0xFFFFFFFFull);
            v = feats[l][((size_t)b * C_ + c) * (size_t)Ps[l] + idx];
        }
        sum += v;
    }
    out[tid] = 0.25f * sum;
}

extern "C" void kernel_launch(void* const* d_in, const int* in_sizes, int n_in,
                              void* d_out, int out_size, void* d_ws, size_t ws_size,
                              hipStream_t stream) {
    const float* feat0 = (const float*)d_in[0];
    const float* feat1 = (const float*)d_in[1];
    const float* feat2 = (const float*)d_in[2];
    const float* feat3 = (const float*)d_in[3];
    const float* scrib = (const float*)d_in[4];
    float* ws = (float*)d_ws;

    float* sA   = ws + S_OFF;
    float* cntA = ws + CNT_OFF;
    unsigned long long* pkA = (unsigned long long*)((char*)d_ws + PK_BYTEOFF);

    // Zero s + cnt + packed (contiguous region; packed zeroed as float pairs).
    int nz = (B_ * I_ * C_ * 4) + (B_ * I_ * 4) + (B_ * I_ * 4 * 2);
    zero_kernel<<<(nz + 255) / 256, 256, 0, stream>>>(sA, nz);

    const float* feats[4] = {feat0, feat1, feat2, feat3};
    const int H[4]  = {128, 64, 32, 16};
    const int CH[4] = {1024, 512, 256, 128};   // K-chunk per wave

    for (int l = 0; l < 4; ++l) {
        int h = H[l], P = h * h;
        int factor = 512 / h;
        int offs = factor / 2 - 1;
        int blocksPerBI = P / 256;
        mask_kernel<<<B_ * I_ * blocksPerBI, 256, 0, stream>>>(
            scrib, ws + SEL_OFF[l], cntA + l * (B_ * I_), pkA + l * (B_ * I_),
            h, factor, offs, blocksPerBI);
    }
    for (int l = 0; l < 4; ++l) {
        int P = H[l] * H[l];
        int chunk = CH[l];
        int nchunks = P / chunk;
        int waves = B_ * 16 * nchunks;          // b x 16 C-tiles x K-chunks
        wmma_masked_sum_kernel<<<waves * 32 / 256, 256, 0, stream>>>(
            feats[l], ws + SEL_OFF[l], sA + l * (B_ * I_ * C_), P, chunk, nchunks);
    }
    finalize_kernel<<<(B_ * I_ * C_) / 256, 256, 0, stream>>>(
        feat0, feat1, feat2, feat3, sA, cntA, pkA, (float*)d_out);
}